// RegionEncoder_23081154249148
// MI455X (gfx1250) — compile-verified
//
#include <hip/hip_runtime.h>

#define RSZ 11          // REGION_SIZE
#define RAD 5           // REGION_RADIUS
#define EMB 128
#define SEQ_L 512
#define POS_PER_BLK 8   // one wave32 per position, 8 waves / block

typedef float v4f __attribute__((ext_vector_type(4)));

__global__ __launch_bounds__(256)
void region_encoder_kernel(const int* __restrict__ seq,
                           const float* __restrict__ W,
                           const float* __restrict__ U,
                           float* __restrict__ out)
{
    __shared__ int s_tok[POS_PER_BLK + 2 * RAD];                 // 18 window tokens
    __shared__ __align__(16) float s_u[POS_PER_BLK][RSZ][EMB];   // 45 KB staging

    const int tid  = threadIdx.x;
    const int wave = tid >> 5;
    const int lane = tid & 31;

    const int p0 = blockIdx.x * POS_PER_BLK;   // first flat position of this block
    const int b  = p0 / SEQ_L;
    const int l0 = p0 % SEQ_L;

    // Stage the token window for the block's 8 positions (OOB -> pad token 0).
    if (tid < POS_PER_BLK + 2 * RAD) {
        int ls = l0 - RAD + tid;
        int t  = 0;
        if (ls >= 0 && ls < SEQ_L) t = seq[b * SEQ_L + ls];
        s_tok[tid] = t;
    }
    __syncthreads();

    const int p        = p0 + wave;
    const int t_center = s_tok[wave + RAD];

    // LDS byte address of this wave's staging tile for this lane's 16B chunk.
    // (low 32 bits of a flat shared-aperture address == LDS byte offset)
    unsigned lds_base =
        (unsigned)(unsigned long long)(const void*)(&s_u[wave][0][lane * 4]);

    // Fire all 11 U-row gathers asynchronously into LDS (ASYNCcnt path).
    #pragma unroll
    for (int k = 0; k < RSZ; ++k) {
        int tk = s_tok[wave + k];                      // token at window slot k
        long long row = (long long)tk * RSZ + k;       // U row index
        unsigned long long gaddr =
            (unsigned long long)(const void*)(U + row * EMB + lane * 4);
        unsigned laddr = lds_base + (unsigned)(k * EMB * 4);
        asm volatile("global_load_async_to_lds_b128 %0, %1, off"
                     :: "v"(laddr), "v"(gaddr)
                     : "memory");
    }

    // Overlap: regular VGPR load of the center embedding chunk.
    v4f w4 = *(const v4f*)(W + (long long)t_center * EMB + lane * 4);

    // Wait for the async global->LDS transfers issued by this wave.
    asm volatile("s_wait_asynccnt 0x0" ::: "memory");

    const float NEG = -3.402823466e38f;
    v4f acc = {NEG, NEG, NEG, NEG};
    #pragma unroll
    for (int k = 0; k < RSZ; ++k) {
        v4f u4 = *(const v4f*)(&s_u[wave][k][lane * 4]);
        v4f pr = u4 * w4;
        acc.x = fmaxf(acc.x, pr.x);
        acc.y = fmaxf(acc.y, pr.y);
        acc.z = fmaxf(acc.z, pr.z);
        acc.w = fmaxf(acc.w, pr.w);
    }

    if (t_center == 0) {           // mask = (seq != 0)
        acc.x = 0.0f; acc.y = 0.0f; acc.z = 0.0f; acc.w = 0.0f;
    }

    *(v4f*)(out + (long long)p * EMB + lane * 4) = acc;
}

extern "C" void kernel_launch(void* const* d_in, const int* in_sizes, int n_in,
                              void* d_out, int out_size, void* d_ws, size_t ws_size,
                              hipStream_t stream) {
    const int*   seq = (const int*)d_in[0];    // (B, L, 1) int
    const float* W   = (const float*)d_in[1];  // (VOCAB, 128) f32
    const float* U   = (const float*)d_in[2];  // (VOCAB*11, 128) f32
    float*       out = (float*)d_out;          // (B, L, 1, 128) f32

    const int npos = in_sizes[0];              // B*L = 16384
    dim3 grid(npos / POS_PER_BLK);
    region_encoder_kernel<<<grid, 256, 0, stream>>>(seq, W, U, out);
}